// SparseProjector_56143812493760
// MI455X (gfx1250) — compile-verified
//
#include <hip/hip_runtime.h>
#include <stdint.h>

// --- CDNA5 TDM descriptor vector types (clang-23 / therock-10.0 6-arg builtin form) ---
typedef __attribute__((ext_vector_type(4))) unsigned int u32x4;
typedef __attribute__((ext_vector_type(4))) int          i32x4;
typedef __attribute__((ext_vector_type(8))) int          i32x8;

#define TPB   256   // 8 waves (wave32) per block
#define CHUNK 4096  // floats per TDM store = 16 KB LDS tile

// Kernel 1: inverse-index table init: inv[p] = -1 for all pixels (vectorized).
__global__ void sp_init_inv(int* __restrict__ inv, int n4) {
  int i = blockIdx.x * blockDim.x + threadIdx.x;
  if (i < n4) {
    ((i32x4*)inv)[i] = (i32x4){-1, -1, -1, -1};
  }
}

// Kernel 2: inv[active_indices[j]] = j (indices unique -> plain stores suffice).
__global__ void sp_build_inv(const int* __restrict__ act, int* __restrict__ inv, int na) {
  int j = blockIdx.x * blockDim.x + threadIdx.x;
  if (j < na) inv[act[j]] = j;
}

// Kernel 3: gather-formulated scatter. Each block produces one contiguous
// CHUNK-wide slice of one batch row in LDS, then DMAs it to HBM with the
// Tensor Data Mover (tensor_store_from_lds), keeping the 1 GiB output write
// fully streaming while tau/w/inv gathers stay L2-resident (192 MB L2 holds
// the whole 64 MiB tau matrix + 4 MiB inverse table).
__global__ void sp_gather_store(const float* __restrict__ tau,
                                const float* __restrict__ w,
                                const int* __restrict__ inv,
                                float* __restrict__ out,
                                int n_active, int total_pixels) {
  __shared__ float buf[CHUNK];

  const int b  = blockIdx.y;
  const int p0 = blockIdx.x * CHUNK;
  const float* taub = tau + (long long)b * n_active;

#pragma unroll
  for (int k = 0; k < CHUNK / TPB; ++k) {
    const int pi = threadIdx.x + k * TPB;      // coalesced within the chunk
    const int j  = inv[p0 + pi];               // L2-resident (4 MiB table)
    float v = 0.0f;
    if (j >= 0) v = taub[j] * w[j];            // random reads, L2-resident (64 MiB)
    buf[pi] = v;
  }
  __syncthreads();

  // Wave 0 issues one TDM store of the whole 16 KB tile: LDS -> global.
  if (threadIdx.x < 32) {
    const unsigned long long gaddr =
        (unsigned long long)(uintptr_t)(out + (long long)b * total_pixels + p0);
    const unsigned ldsoff = (unsigned)(uintptr_t)(void*)buf;  // LDS byte offset = addr[31:0]

    // D# group 0: count=1 | lds_addr | global_addr[56:0] | type=2 ("image")
    u32x4 g0;
    g0[0] = 1u;                                                  // count=1, user mode
    g0[1] = ldsoff;                                              // lds_addr
    g0[2] = (unsigned)gaddr;                                     // global_addr[31:0]
    g0[3] = ((unsigned)(gaddr >> 32) & 0x01FFFFFFu) | (2u << 30);// global_addr[56:32] | type=2

    // D# group 1: data_size=4B; 1-row tile: tensor_dim0 = tile_dim0 = CHUNK,
    // tensor_dim1 = 1, tensor_dim0_stride = CHUNK. No multicast, no padding.
    i32x8 g1;
    g1[0] = (int)(2u << 16);                                   // data_size = 2 (4 bytes)
    g1[1] = (int)(((unsigned)CHUNK & 0xFFFFu) << 16);          // tensor_dim0 [15:0] @ bits 63:48
    g1[2] = (int)(((unsigned)CHUNK >> 16) | (1u << 16));       // tensor_dim0 [31:16] | tensor_dim1=1
    g1[3] = (int)(((unsigned)CHUNK & 0xFFFFu) << 16);          // tile_dim0 = CHUNK @ bits 127:112
    g1[4] = 0;                                                 // tile_dim1 = tile_dim2 = 0 (unused)
    g1[5] = (int)CHUNK;                                        // tensor_dim0_stride low 32
    g1[6] = 0;
    g1[7] = 0;

    // <=2D tensor: remaining descriptor groups zero-filled.
    i32x4 g2 = (i32x4){0, 0, 0, 0};
    i32x4 g3 = (i32x4){0, 0, 0, 0};
    i32x8 g4 = (i32x8){0, 0, 0, 0, 0, 0, 0, 0};

    __builtin_amdgcn_tensor_store_from_lds(g0, g1, g2, g3, g4, 0);
    __builtin_amdgcn_s_wait_tensorcnt(0);   // s_wait_tensorcnt 0 before LDS reuse/exit
  }
}

extern "C" void kernel_launch(void* const* d_in, const int* in_sizes, int n_in,
                              void* d_out, int out_size, void* d_ws, size_t ws_size,
                              hipStream_t stream) {
  const float* tau = (const float*)d_in[0];   // [batch, n_active] f32
  const float* w   = (const float*)d_in[1];   // [n_active] f32
  const int*   act = (const int*)d_in[2];     // [n_active] i32 (unique)

  const int n_active     = in_sizes[1];
  const int batch        = in_sizes[0] / n_active;
  const int total_pixels = out_size / batch;

  int* inv = (int*)d_ws;                      // 4 MiB inverse-index table

  const int n4 = total_pixels / 4;
  sp_init_inv<<<(n4 + TPB - 1) / TPB, TPB, 0, stream>>>(inv, n4);
  sp_build_inv<<<(n_active + TPB - 1) / TPB, TPB, 0, stream>>>(act, inv, n_active);

  dim3 grid(total_pixels / CHUNK, batch);
  sp_gather_store<<<grid, TPB, 0, stream>>>(tau, w, inv, (float*)d_out,
                                            n_active, total_pixels);
}